// Conv2D_Offset_39187281609287
// MI455X (gfx1250) — compile-verified
//
#include <hip/hip_runtime.h>
#include <stdint.h>

typedef float v4f __attribute__((ext_vector_type(4)));

#define CPW      16     // channels staged per chunk (4 chunks of 16 = 64)
#define SSTRIDE  132    // padded LDS row stride for s (breaks bank conflicts in phase 2)

// One workgroup per (b, h) pixel row:
//   phase 1: async-stage rows y0,y1 for CPW channels at a time into LDS,
//            bilinear-interp into s_lds[64][128]
//   phase 2: stream 73728 floats (128 w * 576 ch9) of output with NT b128 stores
__global__ __launch_bounds__(256) void resample_rep9_kernel(
    const float* __restrict__ x, float* __restrict__ out)
{
    __shared__ float rows[CPW * 2 * 128];     // 16 KB raw row staging (reused per chunk)
    __shared__ float s_lds[64 * SSTRIDE];     // 33 KB resampled values for all 64 channels

    const int tid = threadIdx.x;
    const int bh  = blockIdx.x;               // 0..1023
    const int b   = bh >> 7;
    const int h   = bh & 127;

    const float SC = 127.0f / 128.0f;         // (n-1)/(2*norm), exact in fp32

    // vertical taps (uniform over the block)
    float ph = (float)h * SC;
    int   y0 = (int)floorf(ph);
    int   y1 = min(y0 + 1, 127);
    float wy = ph - (float)y0;

    // horizontal taps for this thread's pixel column
    const int wp = tid & 127;
    const int c0 = tid >> 7;                  // 0 or 1: which channel parity this thread interps
    float pw = (float)wp * SC;
    int   x0 = (int)floorf(pw);
    int   x1 = min(x0 + 1, 127);
    float wx = pw - (float)x0;

    const float* xb = x + (size_t)b * (64 * 128 * 128);
    const uint32_t rows_base = (uint32_t)(uintptr_t)(&rows[0]);   // LDS byte offset (low 32 bits of flat ptr)

    for (int chunk = 0; chunk < 64 / CPW; ++chunk) {
        const int cbase = chunk * CPW;

        // ---- async-stage 2 source rows x CPW channels (b128 granularity) ----
#pragma unroll
        for (int i = 0; i < (CPW * 2 * 32) / 256; ++i) {
            int f    = tid + i * 256;           // 0 .. CPW*64-1  (float4 index in staging)
            int cl   = f >> 6;                  // local channel
            int rem  = f & 63;
            int yrow = (rem & 32) ? y1 : y0;    // which source row
            int w4   = rem & 31;                // float4 within row
            int gofs = (cbase + cl) * 65536 + yrow * 512 + w4 * 16;   // byte offset from xb
            uint32_t ldso = rows_base + (uint32_t)f * 16u;            // LDS byte address
            // GVS mode: addr = SGPR64(xb) + VADDR[31:0]; LDS dest offset in VDST VGPR
            asm volatile("global_load_async_to_lds_b128 %0, %1, %2"
                         :: "v"(ldso), "v"(gofs), "s"(xb)
                         : "memory");
        }
        asm volatile("s_wait_asynccnt 0" ::: "memory");  // my async LDS writes landed
        __syncthreads();                                  // everyone's landed

        // ---- bilinear from staged rows -> s_lds ----
#pragma unroll
        for (int cl = c0; cl < CPW; cl += 2) {
            const float* r0 = &rows[(cl * 2 + 0) * 128];
            const float* r1 = &rows[(cl * 2 + 1) * 128];
            float v00 = r0[x0], v01 = r0[x1];
            float v10 = r1[x0], v11 = r1[x1];
            float top = v00 * (1.0f - wx) + v01 * wx;
            float bot = v10 * (1.0f - wx) + v11 * wx;
            s_lds[(cbase + cl) * SSTRIDE + wp] = top * (1.0f - wy) + bot * wy;
        }
        __syncthreads();   // rows reused next chunk; also publishes s_lds after last chunk
    }

    // ---- phase 2: stream out[b][h*73728 + w*576 + 9c + k] = s[c][w] ----
    // 73728 floats = 18432 float4 per WG = 72 NT b128 stores per thread, fully coalesced.
    float* ob = out + (size_t)b * 9437184 + (size_t)h * 73728;
#pragma unroll 4
    for (int j = 0; j < 72; ++j) {
        unsigned q    = (unsigned)tid + (unsigned)j * 256u;  // float4 index in the row block
        unsigned wpix = q / 144u;                            // pixel column (576/4 = 144)
        unsigned e0   = (q - wpix * 144u) * 4u;              // ch9 of lane's first element
        v4f v;
        v.x = s_lds[((e0 + 0u) / 9u) * SSTRIDE + wpix];
        v.y = s_lds[((e0 + 1u) / 9u) * SSTRIDE + wpix];
        v.z = s_lds[((e0 + 2u) / 9u) * SSTRIDE + wpix];
        v.w = s_lds[((e0 + 3u) / 9u) * SSTRIDE + wpix];
        __builtin_nontemporal_store(v, (v4f*)(ob + (size_t)q * 4u));
    }
}

extern "C" void kernel_launch(void* const* d_in, const int* in_sizes, int n_in,
                              void* d_out, int out_size, void* d_ws, size_t ws_size,
                              hipStream_t stream) {
    (void)in_sizes; (void)n_in; (void)d_ws; (void)ws_size; (void)out_size;
    const float* x  = (const float*)d_in[0];   // (8, 64, 128, 128) f32
    float* out      = (float*)d_out;           // (8, 64, 384, 384) f32
    // one workgroup per (b, h): 8 * 128 = 1024 blocks of 256 threads (8 waves each)
    resample_rep9_kernel<<<1024, 256, 0, stream>>>(x, out);
}